// GGCL_F_3882650436606
// MI455X (gfx1250) — compile-verified
//
#include <hip/hip_runtime.h>
#include <hip/hip_bf16.h>

// Problem constants (from reference)
#define NN    50000
#define INF_  256
#define OUTF  64
#define GAMMA 1.0f

typedef float v2f __attribute__((ext_vector_type(2)));
typedef float v8f __attribute__((ext_vector_type(8)));

// ---------------------------------------------------------------------------
// Kernel 0: zero the output (atomics accumulate; harness poisons with 0xAA)
// ---------------------------------------------------------------------------
__global__ void zero_f4_kernel(float4* __restrict__ p, int n4) {
    int i = blockIdx.x * blockDim.x + threadIdx.x;
    if (i < n4) p[i] = make_float4(0.f, 0.f, 0.f, 0.f);
}

// ---------------------------------------------------------------------------
// Kernel 1: fused dual-GEMM + activation epilogue using V_WMMA_F32_16X16X4_F32
//   tabA[n,f] = elu(F@Wm) * exp(-relu(F@Ws))
//   tabB[n,f] = relu(F@Ws) * exp(-relu(F@Ws))^2
// One wave per 16-row tile; 4 col-tiles x 2 matrices = 8 accumulators.
// Weights staged in LDS (128 KB) in K-pair layout -> ds_load_b64 per B frag.
// ---------------------------------------------------------------------------
__global__ __launch_bounds__(256)
void gemm_act_kernel(const float* __restrict__ F,
                     const float* __restrict__ Wm,
                     const float* __restrict__ Ws,
                     float* __restrict__ tabA,
                     float* __restrict__ tabB)
{
    // lds layout: mat*16384 + (k>>1)*128 + n*2 + (k&1)   (k in [0,256), n in [0,64))
    extern __shared__ float lds[];
    const int tid = threadIdx.x;

    for (int idx = tid; idx < INF_ * OUTF; idx += 256) {
        int k = idx >> 6;        // row of W
        int n = idx & 63;        // col of W
        int l = ((k >> 1) << 7) + (n << 1) + (k & 1);
        lds[l]          = Wm[idx];
        lds[16384 + l]  = Ws[idx];
    }
    __syncthreads();

    const int wave    = tid >> 5;
    const int lane    = tid & 31;
    const int rowTile = blockIdx.x * 8 + wave;
    if (rowTile >= (NN / 16)) return;     // uniform per wave -> EXEC all-ones in WMMA

    const int half = lane >> 4;   // 0: K pair {0,1}, 1: K pair {2,3}
    const int ln   = lane & 15;   // M (for A) / N (for B,C)
    const int row0 = rowTile * 16;

    v8f accM0 = {}, accM1 = {}, accM2 = {}, accM3 = {};
    v8f accS0 = {}, accS1 = {}, accS2 = {}, accS3 = {};

    // A fragment base: row = row0 + ln, K offset = half*2
    const float* aptr = F + (size_t)(row0 + ln) * INF_ + half * 2;

    for (int k0 = 0; k0 < INF_; k0 += 4) {
        v2f a = *(const v2f*)(aptr + k0);
        const float* bbase = lds + ((k0 >> 1) + half) * 128;   // + n*2
        #define STEP(T, AM, AS)                                                        \
        {                                                                              \
            int col = (T) * 16 + ln;                                                   \
            v2f bm = *(const v2f*)(bbase + col * 2);                                   \
            v2f bs = *(const v2f*)(bbase + 16384 + col * 2);                           \
            AM = __builtin_amdgcn_wmma_f32_16x16x4_f32(false, a, false, bm,            \
                                                       (short)0, AM, false, false);    \
            AS = __builtin_amdgcn_wmma_f32_16x16x4_f32(false, a, false, bs,            \
                                                       (short)0, AS, false, false);    \
        }
        STEP(0, accM0, accS0)
        STEP(1, accM1, accS1)
        STEP(2, accM2, accS2)
        STEP(3, accM3, accS3)
        #undef STEP
    }

    // Epilogue: C/D layout -> VGPR v holds row (v + 8*half), col = ln (+16*t)
    v8f accM[4] = {accM0, accM1, accM2, accM3};
    v8f accS[4] = {accS0, accS1, accS2, accS3};
    #pragma unroll
    for (int t = 0; t < 4; ++t) {
        #pragma unroll
        for (int v = 0; v < 8; ++v) {
            int row = row0 + half * 8 + v;
            int col = t * 16 + ln;
            float x = accM[t][v];
            float y = accS[t][v];
            float m   = x > 0.f ? x : (__expf(x) - 1.f);   // elu
            float s   = y > 0.f ? y : 0.f;                 // relu
            float att = __expf(-GAMMA * s);
            tabA[row * OUTF + col] = m * att;
            tabB[row * OUTF + col] = s * att * att;
        }
    }
}

// ---------------------------------------------------------------------------
// Kernel 2: edge gather/scale/scatter-add. Tables + outputs are L2-resident
// (12.8 MB each vs 192 MB L2). One thread per (edge, feature-pair).
// ---------------------------------------------------------------------------
__global__ __launch_bounds__(256)
void edge_kernel(const long long* __restrict__ esrc,
                 const long long* __restrict__ edst,
                 const float* __restrict__ a1,
                 const float* __restrict__ a2,
                 const float* __restrict__ tabA,
                 const float* __restrict__ tabB,
                 float* __restrict__ outM,
                 float* __restrict__ outS,
                 int nPairs)
{
    int p = blockIdx.x * blockDim.x + threadIdx.x;
    if (p >= nPairs) return;
    int e = p >> 5;
    int f = (p & 31) << 1;
    int src = (int)esrc[e];
    int dst = (int)edst[e];
    float w1 = a1[e];
    float w2 = a2[e];
    float2 vm = *(const float2*)(tabA + src * OUTF + f);
    float2 vs = *(const float2*)(tabB + src * OUTF + f);
    float* pm = outM + dst * OUTF + f;
    float* ps = outS + dst * OUTF + f;
    unsafeAtomicAdd(pm + 0, w1 * vm.x);
    unsafeAtomicAdd(pm + 1, w1 * vm.y);
    unsafeAtomicAdd(ps + 0, w2 * vs.x);
    unsafeAtomicAdd(ps + 1, w2 * vs.y);
}

// ---------------------------------------------------------------------------
extern "C" void kernel_launch(void* const* d_in, const int* in_sizes, int n_in,
                              void* d_out, int out_size, void* d_ws, size_t ws_size,
                              hipStream_t stream) {
    const float*     F    = (const float*)d_in[0];
    const long long* esrc = (const long long*)d_in[1];   // int64 per reference
    const long long* edst = (const long long*)d_in[2];
    const float*     a1   = (const float*)d_in[3];
    const float*     a2   = (const float*)d_in[4];
    const float*     Wm   = (const float*)d_in[5];
    const float*     Ws   = (const float*)d_in[6];

    float* out  = (float*)d_out;                 // [miu_out | sigma_out]
    float* tabA = (float*)d_ws;                  // miu*att       [N,64]
    float* tabB = tabA + (size_t)NN * OUTF;      // sigma*att^2   [N,64]
    const int E = in_sizes[1];

    // 1) zero outputs (6.4M floats)
    int n4 = out_size / 4;
    zero_f4_kernel<<<(n4 + 255) / 256, 256, 0, stream>>>((float4*)out, n4);

    // 2) fused GEMM + activations -> scratch tables
    int rowTiles = NN / 16;                       // 3125 (exact)
    int gblocks  = (rowTiles + 7) / 8;            // 8 waves/block
    gemm_act_kernel<<<gblocks, 256, 2 * INF_ * OUTF * sizeof(float), stream>>>(
        F, Wm, Ws, tabA, tabB);

    // 3) edge scatter: E*32 threads (float2 per thread over 64 features)
    long long nPairs = (long long)E * 32;
    int eblocks = (int)((nPairs + 255) / 256);
    edge_kernel<<<eblocks, 256, 0, stream>>>(esrc, edst, a1, a2, tabA, tabB,
                                             out, out + (size_t)NN * OUTF,
                                             (int)nPairs);
}